// GCNEdge2Cluster_38732015075723
// MI455X (gfx1250) — compile-verified
//
#include <hip/hip_runtime.h>
#include <hip/hip_bf16.h>

typedef __attribute__((ext_vector_type(2))) float v2f;
typedef __attribute__((ext_vector_type(8))) float v8f;

#define HIDC 64
#define NCL 30
#define NCLP 32
#define REG_W 0.01f

// ---------------- init: deg = 1 (self loop), zero reducers ----------------
__global__ void k_init(float* __restrict__ deg, float* __restrict__ colsum,
                       float* __restrict__ loss_sum, int N) {
  int i = blockIdx.x * blockDim.x + threadIdx.x;
  if (i < N) deg[i] = 1.0f;
  if (i < NCLP) colsum[i] = 0.0f;
  if (i == 0) *loss_sum = 0.0f;
}

// ---------------- deg[dst] += w ----------------
__global__ void k_deg_accum(const long long* __restrict__ dst,
                            const float* __restrict__ w,
                            float* __restrict__ deg, int E) {
  int e = blockIdx.x * blockDim.x + threadIdx.x;
  if (e < E) atomicAdd(&deg[dst[e]], w[e]);
}

__global__ void k_dinv(const float* __restrict__ deg, float* __restrict__ dinv, int N) {
  int i = blockIdx.x * blockDim.x + threadIdx.x;
  if (i < N) dinv[i] = rsqrtf(deg[i]);
}

// ---------------- GEMM1: h = x @ W1  (x:[N,64], W1:[64,64]) via f32 WMMA ----------------
// One wave per 16x16 tile; 4 waves/block cover the 4 N-tiles of one M-tile.
__global__ void k_gemm1(const float* __restrict__ x, const float* __restrict__ W1,
                        float* __restrict__ h, int N) {
  const int lane = threadIdx.x & 31;
  const int wave = threadIdx.x >> 5;          // 0..3 -> n tile
  const int m0 = blockIdx.x * 16;
  const int n0 = wave * 16;
  int mrow = m0 + (lane & 15);
  if (mrow >= N) mrow = N - 1;                // clamp loads; EXEC stays full for WMMA
  const int nn = n0 + (lane & 15);
  const int kh = (lane >> 4) * 2;             // ISA A/B layout: lanes16-31 hold K+2
  v8f acc = {};
#pragma unroll
  for (int k = 0; k < HIDC; k += 4) {
    v2f a, b;
    a[0] = x[(size_t)mrow * HIDC + k + kh];
    a[1] = x[(size_t)mrow * HIDC + k + kh + 1];
    b[0] = W1[(size_t)(k + kh) * HIDC + nn];
    b[1] = W1[(size_t)(k + kh + 1) * HIDC + nn];
    acc = __builtin_amdgcn_wmma_f32_16x16x4_f32(false, a, false, b,
                                                (short)0, acc, false, false);
  }
  const int mbase = m0 + ((lane >> 4) * 8);   // D: VGPR r -> row r (+8 for hi lanes)
  const int nc = n0 + (lane & 15);
#pragma unroll
  for (int r = 0; r < 8; ++r) {
    int m = mbase + r;
    if (m < N) h[(size_t)m * HIDC + nc] = acc[r];
  }
}

// ---------------- agg1 = h/deg + b1 (self-loop + bias) ----------------
__global__ void k_agg_init1(const float* __restrict__ h, const float* __restrict__ deg,
                            const float* __restrict__ b1, float* __restrict__ agg, int N) {
  int i = blockIdx.x * blockDim.x + threadIdx.x;
  if (i < N * HIDC) {
    int node = i >> 6;
    agg[i] = h[i] / deg[node] + b1[i & 63];
  }
}

// ---------------- scatter: agg1[dst] += dinv[s]*w*dinv[d] * h[src]  (wave per edge) ----------------
__global__ void k_scatter1(const long long* __restrict__ src, const long long* __restrict__ dst,
                           const float* __restrict__ w, const float* __restrict__ dinv,
                           const float* __restrict__ h, float* __restrict__ agg, int E) {
  int e = blockIdx.x * (256 / 32) + (threadIdx.x >> 5);
  if (e >= E) return;
  int lane = threadIdx.x & 31;
  long long s = src[e], d = dst[e];
  float coeff = dinv[s] * w[e] * dinv[d];
  const float* hs = h + (size_t)s * HIDC;
  float* ad = agg + (size_t)d * HIDC;
  atomicAdd(&ad[lane],        coeff * hs[lane]);
  atomicAdd(&ad[lane + 32],   coeff * hs[lane + 32]);
}

__global__ void k_relu(float* __restrict__ a, int n) {
  int i = blockIdx.x * blockDim.x + threadIdx.x;
  if (i < n) a[i] = fmaxf(a[i], 0.0f);
}

// ---------------- GEMM2: h2 = h1 @ W2  (W2:[64,30], N padded to 32) ----------------
__global__ void k_gemm2(const float* __restrict__ h1, const float* __restrict__ W2,
                        float* __restrict__ h2, int N) {
  const int lane = threadIdx.x & 31;
  const int wave = threadIdx.x >> 5;          // 0..1 -> n tile
  const int m0 = blockIdx.x * 16;
  const int n0 = wave * 16;
  int mrow = m0 + (lane & 15);
  if (mrow >= N) mrow = N - 1;
  const int nn = n0 + (lane & 15);
  const int kh = (lane >> 4) * 2;
  const bool nvalid = (nn < NCL);
  v8f acc = {};
#pragma unroll
  for (int k = 0; k < HIDC; k += 4) {
    v2f a, b;
    a[0] = h1[(size_t)mrow * HIDC + k + kh];
    a[1] = h1[(size_t)mrow * HIDC + k + kh + 1];
    b[0] = nvalid ? W2[(size_t)(k + kh) * NCL + nn] : 0.0f;
    b[1] = nvalid ? W2[(size_t)(k + kh + 1) * NCL + nn] : 0.0f;
    acc = __builtin_amdgcn_wmma_f32_16x16x4_f32(false, a, false, b,
                                                (short)0, acc, false, false);
  }
  const int mbase = m0 + ((lane >> 4) * 8);
  const int nc = n0 + (lane & 15);
#pragma unroll
  for (int r = 0; r < 8; ++r) {
    int m = mbase + r;
    if (m < N) h2[(size_t)m * NCLP + nc] = acc[r];
  }
}

// ---------------- agg2 = h2/deg + b2 (cols<30), 0 in pad cols ----------------
__global__ void k_agg_init2(const float* __restrict__ h2, const float* __restrict__ deg,
                            const float* __restrict__ b2, float* __restrict__ agg, int N) {
  int i = blockIdx.x * blockDim.x + threadIdx.x;
  if (i < N * NCLP) {
    int node = i >> 5, c = i & 31;
    agg[i] = (c < NCL) ? (h2[i] / deg[node] + b2[c]) : 0.0f;
  }
}

__global__ void k_scatter2(const long long* __restrict__ src, const long long* __restrict__ dst,
                           const float* __restrict__ w, const float* __restrict__ dinv,
                           const float* __restrict__ h2, float* __restrict__ agg, int E) {
  int e = blockIdx.x * (256 / 32) + (threadIdx.x >> 5);
  if (e >= E) return;
  int lane = threadIdx.x & 31;
  long long s = src[e], d = dst[e];
  float coeff = dinv[s] * w[e] * dinv[d];
  if (lane < NCL) {
    float v = coeff * h2[(size_t)s * NCLP + lane];
    atomicAdd(&agg[(size_t)d * NCLP + lane], v);
  }
}

// ---------------- softmax rows -> FX (d_out), plus colsum of log(1-FX^2) ----------------
__global__ void k_softmax(const float* __restrict__ agg2, float* __restrict__ FX,
                          float* __restrict__ colsum, int N) {
  __shared__ float scol[NCL];
  int t = threadIdx.x;
  if (t < NCL) scol[t] = 0.0f;
  __syncthreads();
  int i = blockIdx.x * blockDim.x + t;
  if (i < N) {
    float v[NCL];
    float mx = -3.402823466e+38f;
#pragma unroll
    for (int j = 0; j < NCL; ++j) { v[j] = agg2[(size_t)i * NCLP + j]; mx = fmaxf(mx, v[j]); }
    float sum = 0.0f;
#pragma unroll
    for (int j = 0; j < NCL; ++j) { v[j] = expf(v[j] - mx); sum += v[j]; }
    float inv = 1.0f / sum;
#pragma unroll
    for (int j = 0; j < NCL; ++j) {
      float f = v[j] * inv;
      FX[(size_t)i * NCL + j] = f;
      atomicAdd(&scol[j], logf(1.0f - f * f));
    }
  }
  __syncthreads();
  if (t < NCL) atomicAdd(&colsum[t], scol[t]);
}

// ---------------- loss: mean((FF - w)^2), FF = <FX[src], FX[dst]> ----------------
__global__ void k_loss(const long long* __restrict__ src, const long long* __restrict__ dst,
                       const float* __restrict__ w, const float* __restrict__ FX,
                       float* __restrict__ loss_sum, int E) {
  __shared__ float red[256];
  int t = threadIdx.x;
  int e = blockIdx.x * blockDim.x + t;
  float se = 0.0f;
  if (e < E) {
    const float* a = FX + (size_t)src[e] * NCL;
    const float* b = FX + (size_t)dst[e] * NCL;
    float ff = 0.0f;
#pragma unroll
    for (int j = 0; j < NCL; ++j) ff += a[j] * b[j];
    float d = ff - w[e];
    se = d * d;
  }
  red[t] = se;
  __syncthreads();
  for (int s = 128; s > 0; s >>= 1) {
    if (t < s) red[t] += red[t + s];
    __syncthreads();
  }
  if (t == 0) atomicAdd(loss_sum, red[0]);
}

__global__ void k_finalize(const float* __restrict__ colsum, const float* __restrict__ loss_sum,
                           float* __restrict__ out_loss, int E) {
  float preg = 0.0f;
#pragma unroll
  for (int j = 0; j < NCL; ++j) preg -= logf(1.0001f - expf(colsum[j]));
  *out_loss = (*loss_sum) / (float)E + REG_W * preg;
}

extern "C" void kernel_launch(void* const* d_in, const int* in_sizes, int n_in,
                              void* d_out, int out_size, void* d_ws, size_t ws_size,
                              hipStream_t stream) {
  const float*     x  = (const float*)d_in[0];
  const long long* ei = (const long long*)d_in[1];
  const float*     ew = (const float*)d_in[2];
  const float*     W1 = (const float*)d_in[3];
  const float*     b1 = (const float*)d_in[4];
  const float*     W2 = (const float*)d_in[5];
  const float*     b2 = (const float*)d_in[6];

  const int N = in_sizes[0] / HIDC;     // 100000
  const int E = in_sizes[2];            // 1000000
  const long long* src = ei;
  const long long* dst = ei + E;

  float* ws       = (float*)d_ws;
  float* deg      = ws;                                   // N
  float* dinv     = ws + (size_t)N;                       // N
  float* bufH     = ws + 2 * (size_t)N;                   // N*64 (h1; then h2/agg2)
  float* agg1     = bufH + (size_t)N * HIDC;              // N*64
  float* colsum   = agg1 + (size_t)N * HIDC;              // 32
  float* loss_sum = colsum + NCLP;                        // 1
  float* h2       = bufH;                                 // N*32 (reuse)
  float* agg2     = bufH + (size_t)N * NCLP;              // N*32 (reuse)

  float* FX  = (float*)d_out;                             // N*30
  float* out_loss = FX + (size_t)N * NCL;                 // +1

  const int gN   = (N + 255) / 256;
  const int gE   = (E + 255) / 256;
  const int gEw  = (E + 7) / 8;                           // wave-per-edge, 8 waves/block
  const int gN64 = ((size_t)N * HIDC + 255) / 256;
  const int gN32 = ((size_t)N * NCLP + 255) / 256;
  const int gMt  = (N + 15) / 16;

  k_init      <<<gN,   256, 0, stream>>>(deg, colsum, loss_sum, N);
  k_deg_accum <<<gE,   256, 0, stream>>>(dst, ew, deg, E);
  k_dinv      <<<gN,   256, 0, stream>>>(deg, dinv, N);

  k_gemm1     <<<gMt,  128, 0, stream>>>(x, W1, bufH, N);
  k_agg_init1 <<<gN64, 256, 0, stream>>>(bufH, deg, b1, agg1, N);
  k_scatter1  <<<gEw,  256, 0, stream>>>(src, dst, ew, dinv, bufH, agg1, E);
  k_relu      <<<gN64, 256, 0, stream>>>(agg1, N * HIDC);

  k_gemm2     <<<gMt,   64, 0, stream>>>(agg1, W2, h2, N);
  k_agg_init2 <<<gN32, 256, 0, stream>>>(h2, deg, b2, agg2, N);
  k_scatter2  <<<gEw,  256, 0, stream>>>(src, dst, ew, dinv, h2, agg2, E);

  k_softmax   <<<gN,   256, 0, stream>>>(agg2, FX, colsum, N);
  k_loss      <<<gE,   256, 0, stream>>>(src, dst, ew, FX, loss_sum, E);
  k_finalize  <<<1,      1, 0, stream>>>(colsum, loss_sum, out_loss, E);
}